// RayTracing_52286931861554
// MI455X (gfx1250) — compile-verified
//
#include <hip/hip_runtime.h>

typedef __attribute__((ext_vector_type(2))) float v2f;
typedef __attribute__((ext_vector_type(4))) float v4f;
typedef __attribute__((ext_vector_type(8))) float v8f;

#define N_RAYS    65536
#define N_SAMPLES 64
#define NPTS      (N_RAYS * N_SAMPLES)   // 4,194,304 points
#define HIDDEN    128
#define PTS_FLOATS (NPTS * 3)            // 12,582,912 floats (sampled_points copy)

// ---------------------------------------------------------------------------
// Kernel 1: pass-through copy of samples -> d_out[0 .. PTS_FLOATS)
// ---------------------------------------------------------------------------
__global__ __launch_bounds__(256) void copy_samples_kernel(
    const v4f* __restrict__ in, v4f* __restrict__ out, int n4)
{
    int i = blockIdx.x * 256 + threadIdx.x;
    if (i < n4) out[i] = in[i];
}

// ---------------------------------------------------------------------------
// Kernel 2: fused MLP  (3 -> 128 relu -> 128 relu -> 1)
// One wave32 handles 16 points. Layer-2 GEMM runs on V_WMMA_F32_16X16X4_F32
// with W2 staged in LDS pre-swizzled into the B-operand lane layout.
// ---------------------------------------------------------------------------
__global__ __launch_bounds__(256) void sdf_mlp_kernel(
    const float* __restrict__ samples,   // [NPTS,3]
    const float* __restrict__ W1,        // [3,128]
    const float* __restrict__ b1,        // [128]
    const float* __restrict__ W2,        // [128,128]
    const float* __restrict__ b2,        // [128]
    const float* __restrict__ W3,        // [128,1]
    const float* __restrict__ b3,        // [1]
    float* __restrict__ out_net)         // [NPTS]
{
    // --- LDS staging ---------------------------------------------------------
    // w2p[(kc*2+hi)*128 + col] = { W2[4kc+2hi][col], W2[4kc+2hi+1][col] }
    __shared__ v2f  w2p[8192];        // 64 KB
    __shared__ v4f  w1b1[HIDDEN];     // per-col {W1[0][c],W1[1][c],W1[2][c],b1[c]}
    __shared__ float b2s[HIDDEN];
    __shared__ float w3s[HIDDEN];

    const int tid = threadIdx.x;

    // stage W2 swizzled into B-operand pair layout (coalesced global reads)
    for (int e = tid; e < HIDDEN * HIDDEN; e += 256) {
        int k   = e >> 7;            // row of W2
        int col = e & 127;           // column
        int kc  = k >> 2;            // K-chunk of 4
        int hi  = (k >> 1) & 1;      // which lane-half holds this pair
        int slot = k & 1;            // element within the pair
        w2p[(kc * 2 + hi) * HIDDEN + col][slot] = W2[e];
    }
    // stage W1 + b1 packed per column
    for (int e = tid; e < HIDDEN * 4; e += 256) {
        int c = e >> 2, k = e & 3;
        w1b1[c][k] = (k < 3) ? W1[k * HIDDEN + c] : b1[c];
    }
    for (int e = tid; e < HIDDEN; e += 256) {
        b2s[e] = b2[e];
        w3s[e] = W3[e];
    }
    __syncthreads();

    // --- per-wave tile of 16 points -----------------------------------------
    const int lane = tid & 31;
    const int ln   = lane & 15;      // row M within tile / column sub-index
    const int hi   = lane >> 4;      // lane half (selects K pair / M+8)
    const long tileBase = ((long)blockIdx.x * 8 + (tid >> 5)) * 16;

    // layer 1: compute h1 row M=ln directly into WMMA A-operand layout.
    // A 16x4 f32: vgpr0 = K (hi?2:0), vgpr1 = K (hi?3:1) of row M=ln.
    const long pidx = (tileBase + ln) * 3;
    const float px = samples[pidx + 0];
    const float py = samples[pidx + 1];
    const float pz = samples[pidx + 2];

    v2f a[32];
#pragma unroll
    for (int kc = 0; kc < 32; ++kc) {
        int j0 = kc * 4 + hi * 2;
        v4f q0 = w1b1[j0];
        v4f q1 = w1b1[j0 + 1];
        float h0 = px * q0.x + py * q0.y + pz * q0.z + q0.w;
        float h1 = px * q1.x + py * q1.y + pz * q1.z + q1.w;
        a[kc].x = h0 > 0.f ? h0 : 0.f;
        a[kc].y = h1 > 0.f ? h1 : 0.f;
    }

    // layers 2+3: 8 N-tiles of 16 columns; fold the 128->1 dot into each tile.
    float s[8];
#pragma unroll
    for (int r = 0; r < 8; ++r) s[r] = 0.f;

    const float b3v = b3[0];

    for (int nt = 0; nt < 8; ++nt) {           // rolled: 32 static WMMAs
        const int col  = nt * 16 + ln;
        const int base = hi * HIDDEN + col;    // into w2p, + kc*256 per step
        const float bias = b2s[col];

        v8f acc;
#pragma unroll
        for (int r = 0; r < 8; ++r) acc[r] = bias;

#pragma unroll
        for (int kc = 0; kc < 32; ++kc) {
            v2f bop = w2p[kc * 256 + base];
            acc = __builtin_amdgcn_wmma_f32_16x16x4_f32(
                false, a[kc], false, bop, (short)0, acc, false, false);
        }

        // relu + layer-3 partial dot: lane holds C[M=r+hi*8][col]
        const float w3v = w3s[col];
#pragma unroll
        for (int r = 0; r < 8; ++r) {
            float h = acc[r] > 0.f ? acc[r] : 0.f;
            s[r] += h * w3v;
        }
    }

    // reduce the partial dots across the 16 lanes of each half-wave
#pragma unroll
    for (int r = 0; r < 8; ++r) {
        float v = s[r];
        v += __shfl_xor(v, 1, 32);
        v += __shfl_xor(v, 2, 32);
        v += __shfl_xor(v, 4, 32);
        v += __shfl_xor(v, 8, 32);
        s[r] = v;
    }

    if (ln == 0) {                       // lane 0 -> M=0..7, lane 16 -> M=8..15
        long outBase = tileBase + (long)hi * 8;
#pragma unroll
        for (int r = 0; r < 8; ++r) out_net[outBase + r] = s[r] + b3v;
    }
}

// ---------------------------------------------------------------------------
// launcher
// ---------------------------------------------------------------------------
extern "C" void kernel_launch(void* const* d_in, const int* in_sizes, int n_in,
                              void* d_out, int out_size, void* d_ws, size_t ws_size,
                              hipStream_t stream)
{
    const float* samples = (const float*)d_in[0];
    const float* W1      = (const float*)d_in[4];
    const float* b1      = (const float*)d_in[5];
    const float* W2      = (const float*)d_in[6];
    const float* b2      = (const float*)d_in[7];
    const float* W3      = (const float*)d_in[8];
    const float* b3      = (const float*)d_in[9];

    float* out = (float*)d_out;

    // output[0 .. PTS_FLOATS) = samples (pass-through)
    int n4 = PTS_FLOATS / 4;
    copy_samples_kernel<<<(n4 + 255) / 256, 256, 0, stream>>>(
        (const v4f*)samples, (v4f*)out, n4);

    // output[PTS_FLOATS ..) = net values
    int nBlocks = NPTS / (16 * 8);   // 16 pts/wave, 8 waves/block = 32768 blocks
    sdf_mlp_kernel<<<nBlocks, 256, 0, stream>>>(
        samples, W1, b1, W2, b2, W3, b3, out + PTS_FLOATS);
}